// HybridPrimitives_39049842655733
// MI455X (gfx1250) — compile-verified
//
#include <hip/hip_runtime.h>

typedef __attribute__((ext_vector_type(2))) float v2f;
typedef __attribute__((ext_vector_type(8))) float v8f;

#define NQ      100000
#define NG      1000
#define NW      1000
#define NC      3
#define NTILES  (NQ / 16)          // 6250 tiles of 16 points, exact
#define WPB     8                  // waves per block (wave32)
#define BLOCK   (WPB * 32)
#define TWO_PI  6.28318530717958647692f

// ---- LDS float offsets. Phase A: gaussian params (36 KB). Phase B: wave
// params (64 KB) reuse the same buffer. Total static LDS = 64000 B. ----
#define GMEAN   0                  // [NG][2]
#define GMAT    2000               // [NG][4]  (16B aligned: 2000*4 = 8000)
#define GCOL    6000               // [NG][3]

#define WMEAN   0                  // [NW][2]
#define WMAT    2000               // [NW][4]
#define WCOL    6000               // [NW][3]
#define WFREQ   9000               // [NW][2]  (9000*4 = 36000, 8B aligned)
#define WCOEF   11000              // [NW][5]  scalar reads
#define LDS_F   16000              // 64000 bytes

__global__ __launch_bounds__(BLOCK) void hybrid_prims_kernel(
    const float* __restrict__ x,
    const float* __restrict__ gcol,
    const float* __restrict__ gmean,
    const float* __restrict__ gmat,
    const float* __restrict__ wcol,
    const float* __restrict__ wmean,
    const float* __restrict__ wmat,
    const float* __restrict__ wfreq,
    const float* __restrict__ wcoef,
    float* __restrict__ out)
{
    __shared__ __align__(16) float lds[LDS_F];

    const int lane = threadIdx.x & 31;
    const int wv   = threadIdx.x >> 5;
    const int tile = blockIdx.x * WPB + wv;
    const bool active = (tile < NTILES);   // wave-uniform

    // A-matrix: lanes 0-15 -> rows M=0..15 with K=0,1 ; lanes 16-31 -> K=2,3.
    // B-matrix: lanes 0-15 -> cols N=0..15 with K=0,1 ; lanes 16-31 -> K=2,3.
    // C/D:      lanes 0-15 -> cols N, rows M=vgpr 0..7 ; lanes 16-31 -> M=8..15.
    const int   p      = lane & 15;          // point row (A) == channel col (B/D)
    const int   koff   = (lane >> 4) << 1;   // 0 or 2
    const float chMask = (p < NC) ? 1.0f : 0.0f;
    const int   cc     = (p < NC) ? p : 0;

    float rx = 0.0f, ry = 0.0f;
    if (active) {
        __builtin_prefetch(&x[tile * 32], 0, 0);     // global_prefetch_b8
        const float2 xy = *reinterpret_cast<const float2*>(&x[(tile * 16 + p) * 2]);
        rx = xy.x; ry = xy.y;
    }

    // ---------------- phase A: stage gaussian params in LDS ----------------
    for (int i = threadIdx.x; i < 2 * NG; i += BLOCK) lds[GMEAN + i] = gmean[i];
    for (int i = threadIdx.x; i < 4 * NG; i += BLOCK) lds[GMAT  + i] = gmat[i];
    for (int i = threadIdx.x; i < 3 * NG; i += BLOCK) lds[GCOL  + i] = gcol[i];
    __syncthreads();

    v8f acc = {};   // fp32 16x16 accumulator, carried across both passes

    if (active) {
        for (int k0 = 0; k0 < NG; k0 += 4) {
            v2f A, B;
            #pragma unroll
            for (int j = 0; j < 2; ++j) {
                const int g = k0 + koff + j;
                const float2 mn = *reinterpret_cast<const float2*>(&lds[GMEAN + g * 2]);
                const float4 M  = *reinterpret_cast<const float4*>(&lds[GMAT  + g * 4]);
                const float dx = rx - mn.x, dy = ry - mn.y;
                const float t0 = dx * M.x + dy * M.z;   // rel @ mat, e=0
                const float t1 = dx * M.y + dy * M.w;   // e=1
                const float w  = __expf(-0.5f * (t0 * t0 + t1 * t1));
                A[j] = w;
                B[j] = chMask * lds[GCOL + g * 3 + cc];
            }
            // D = A(16x4 f32) x B(4x16 f32) + C : accumulates weights @ colors
            acc = __builtin_amdgcn_wmma_f32_16x16x4_f32(
                      false, A, false, B, (short)0, acc, false, false);
        }
    }
    __syncthreads();

    // ---------------- phase B: stage wave params in LDS --------------------
    for (int i = threadIdx.x; i < 2 * NW; i += BLOCK) lds[WMEAN + i] = wmean[i];
    for (int i = threadIdx.x; i < 4 * NW; i += BLOCK) lds[WMAT  + i] = wmat[i];
    for (int i = threadIdx.x; i < 3 * NW; i += BLOCK) lds[WCOL  + i] = wcol[i];
    for (int i = threadIdx.x; i < 2 * NW; i += BLOCK) lds[WFREQ + i] = wfreq[i];
    for (int i = threadIdx.x; i < 5 * NW; i += BLOCK) lds[WCOEF + i] = wcoef[i];
    __syncthreads();

    if (active) {
        for (int k0 = 0; k0 < NW; k0 += 4) {
            v2f A, B;
            #pragma unroll
            for (int j = 0; j < 2; ++j) {
                const int w = k0 + koff + j;
                // super-Gaussian support: exp(-0.5*(t0^10 + t1^10))
                const float2 mn = *reinterpret_cast<const float2*>(&lds[WMEAN + w * 2]);
                const float4 M  = *reinterpret_cast<const float4*>(&lds[WMAT  + w * 4]);
                const float dx = rx - mn.x, dy = ry - mn.y;
                const float t0 = dx * M.x + dy * M.z;
                const float t1 = dx * M.y + dy * M.w;
                const float t0_2 = t0 * t0,     t1_2 = t1 * t1;
                const float t0_4 = t0_2 * t0_2, t1_4 = t1_2 * t1_2;
                const float t0_8 = t0_4 * t0_4, t1_8 = t1_4 * t1_4;
                const float sup  = __expf(-0.5f * (t0_8 * t0_2 + t1_8 * t1_2));
                // 2D Fourier wave
                const float2 fr = *reinterpret_cast<const float2*>(&lds[WFREQ + w * 2]);
                const float px = TWO_PI * rx * fr.x;
                const float py = TWO_PI * ry * fr.y;
                const float sx = __sinf(px), cx = __cosf(px);
                const float sy = __sinf(py), cy = __cosf(py);
                const float c0 = lds[WCOEF + w * 5 + 0];
                const float c1 = lds[WCOEF + w * 5 + 1];
                const float c2 = lds[WCOEF + w * 5 + 2];
                const float c3 = lds[WCOEF + w * 5 + 3];
                const float c4 = lds[WCOEF + w * 5 + 4];
                const float wvv = c0 * sx * sy + c1 * sx * cy
                                + c2 * cx * sy + c3 * cx * cy + c4;
                A[j] = sup * wvv;
                B[j] = chMask * lds[WCOL + w * 3 + cc];
            }
            acc = __builtin_amdgcn_wmma_f32_16x16x4_f32(
                      false, A, false, B, (short)0, acc, false, false);
        }
    }

    // ---------------- write out: lane holds col N=p, rows M=r(+8) ----------
    if (active && p < NC) {
        const int mbase = tile * 16 + ((lane < 16) ? 0 : 8);
        #pragma unroll
        for (int r = 0; r < 8; ++r)
            out[(mbase + r) * NC + p] = acc[r];
    }
}

extern "C" void kernel_launch(void* const* d_in, const int* in_sizes, int n_in,
                              void* d_out, int out_size, void* d_ws, size_t ws_size,
                              hipStream_t stream) {
    (void)in_sizes; (void)n_in; (void)d_ws; (void)ws_size; (void)out_size;
    const float* x     = (const float*)d_in[0];
    const float* gcol  = (const float*)d_in[1];
    const float* gmean = (const float*)d_in[2];
    const float* gmat  = (const float*)d_in[3];
    const float* wcol  = (const float*)d_in[4];
    const float* wmean = (const float*)d_in[5];
    const float* wmat  = (const float*)d_in[6];
    const float* wfreq = (const float*)d_in[7];
    const float* wcoef = (const float*)d_in[8];
    float* out = (float*)d_out;

    const int grid = (NTILES + WPB - 1) / WPB;   // 782 blocks
    hipLaunchKernelGGL(hybrid_prims_kernel, dim3(grid), dim3(BLOCK), 0, stream,
                       x, gcol, gmean, gmat, wcol, wmean, wmat, wfreq, wcoef, out);
}